// HMLDM_37065567765086
// MI455X (gfx1250) — compile-verified
//
#include <hip/hip_runtime.h>
#include <math.h>

#define N1v 8192
#define N2v 8192
#define Dv 16
#define Ev 262144
#define DELTAv 1.0f
#define EPSv 1e-6f
#define NLOG2E (-1.44269504088896340736f)  // -log2(e), for exp(-x) = exp2(-x*log2e)

typedef float v2f __attribute__((ext_vector_type(2)));
typedef float v8f __attribute__((ext_vector_type(8)));

// workspace layout (floats):
//   [0,        N1)            z2   = sum(z*z) per row of Z
//   [N1,       2N1)           eg1  = exp(gamma_n1)
//   [2N1,      2N1+2*N2)      interleaved column pairs: (w2[n], eg2[n]*exp(-DELTA*EPS))
//   [2N1+2N2,  +512)          per-block partials for pdist1
//   [2N1+2N2+512, +512)       per-block partials for pdist2
#define WS_Z2   0
#define WS_EG1  (N1v)
#define WS_WC   (2 * N1v)
#define WS_P1   (2 * N1v + 2 * N2v)
#define WS_P2   (2 * N1v + 2 * N2v + 512)

__global__ __launch_bounds__(256) void precompute_kernel(
    const float* __restrict__ g1, const float* __restrict__ g2,
    const float* __restrict__ z, const float* __restrict__ w,
    float* __restrict__ ws) {
  int i = blockIdx.x * 256 + threadIdx.x;
  if (i < N1v) {
    const float4* zr = (const float4*)(z + (size_t)i * Dv);
    float s = 0.f;
#pragma unroll
    for (int q = 0; q < 4; ++q) {
      float4 v = zr[q];
      s += v.x * v.x + v.y * v.y + v.z * v.z + v.w * v.w;
    }
    ws[WS_Z2 + i] = s;
    ws[WS_EG1 + i] = __expf(g1[i]);
  }
  if (i < N2v) {
    const float4* wr = (const float4*)(w + (size_t)i * Dv);
    float s = 0.f;
#pragma unroll
    for (int q = 0; q < 4; ++q) {
      float4 v = wr[q];
      s += v.x * v.x + v.y * v.y + v.z * v.z + v.w * v.w;
    }
    // interleaved (w2, eg2) pair; EPS constant folded into eg2
    float2 pair;
    pair.x = s;
    pair.y = __expf(g2[i]) * __expf(-DELTAv * EPSv);
    ((float2*)(ws + WS_WC))[i] = pair;
  }
}

// Dense term: one block per 16-row m-tile (512 blocks), 8 waves per block.
// Each wave streams n-tiles with stride 8 and does S = Z_tile @ W_tile^T via
// 4 chained v_wmma_f32_16x16x4_f32 (K = 4*4 = 16, exact fp32).
__global__ __launch_bounds__(256) void pdist1_kernel(
    const float* __restrict__ z, const float* __restrict__ w,
    const float* __restrict__ ws, float* __restrict__ partials) {
  const int m0 = blockIdx.x * 16;
  const int lane = threadIdx.x & 31;
  const int wave = threadIdx.x >> 5;
  const int l = lane & 15;  // row-within-tile index for A/B loads, N index for C
  const int h = lane >> 4;  // half-wave select

  const float* z2  = ws + WS_Z2;
  const float* eg1 = ws + WS_EG1;
  const float2* wc = (const float2*)(ws + WS_WC);  // (w2, eg2) pairs

  // A fragment (Z tile, 16x16 split as four 16x4 chunks), resident in regs.
  // ISA 32-bit A 16x4 layout: lanes 0-15 -> K={0,1}, lanes 16-31 -> K={2,3}.
  v2f a[4];
#pragma unroll
  for (int kk = 0; kk < 4; ++kk) {
    const float* p = z + (size_t)(m0 + l) * Dv + 4 * kk + 2 * h;
    a[kk][0] = p[0];
    a[kk][1] = p[1];
  }

  // Row-side constants for the 8 accumulator rows this lane owns:
  // C VGPR v, lanes 0-15 -> M = v ; lanes 16-31 -> M = v + 8.
  float rz2[8], re1[8];
#pragma unroll
  for (int v = 0; v < 8; ++v) {
    int m = m0 + v + 8 * h;
    rz2[v] = z2[m];
    re1[v] = eg1[m];
  }

  v2f acc2 = {0.f, 0.f};  // packed accumulator -> v_pk_add_f32
  for (int nt = wave; nt < N2v / 16; nt += 8) {
    const int n0 = nt * 16;
    // B fragment: B = W^T, and the 32-bit B(4x16) layout mirrors A's, so the
    // load pattern is identical with W rows in place of Z rows.
    v2f b[4];
#pragma unroll
    for (int kk = 0; kk < 4; ++kk) {
      const float* p = w + (size_t)(n0 + l) * Dv + 4 * kk + 2 * h;
      b[kk][0] = p[0];
      b[kk][1] = p[1];
    }

    v8f c = {};
#pragma unroll
    for (int kk = 0; kk < 4; ++kk) {
      c = __builtin_amdgcn_wmma_f32_16x16x4_f32(
          /*neg_a=*/false, a[kk], /*neg_b=*/false, b[kk],
          /*c_mod=*/(short)0, c, /*reuse_a=*/false, /*reuse_b=*/false);
    }

    // single b64 load for both column constants
    const float2 col = wc[n0 + l];
    const float cw2 = col.x;
    const float ce2 = col.y;
#pragma unroll
    for (int v = 0; v < 8; v += 2) {
      float sq0 = fmaxf(rz2[v]     + cw2 - 2.0f * c[v],     0.0f);
      float sq1 = fmaxf(rz2[v + 1] + cw2 - 2.0f * c[v + 1], 0.0f);
      // raw v_sqrt_f32: inputs are in [0, ~20], no denormal-range fixup needed
      float d0 = __builtin_amdgcn_sqrtf(sq0);
      float d1 = __builtin_amdgcn_sqrtf(sq1);
      // exp(-DELTA*d) as a single v_exp_f32 (EPS already folded into ce2)
      float e0 = __builtin_amdgcn_exp2f(NLOG2E * DELTAv * d0);
      float e1 = __builtin_amdgcn_exp2f(NLOG2E * DELTAv * d1);
      acc2[0] += re1[v]     * (ce2 * e0);
      acc2[1] += re1[v + 1] * (ce2 * e1);
    }
  }

  // wave32 reduction, then block reduction -> deterministic per-block partial
  float acc = acc2[0] + acc2[1];
#pragma unroll
  for (int off = 16; off > 0; off >>= 1) acc += __shfl_down(acc, off, 32);
  __shared__ float smem[8];
  if (lane == 0) smem[wave] = acc;
  __syncthreads();
  if (threadIdx.x == 0) {
    float s = 0.f;
#pragma unroll
    for (int i = 0; i < 8; ++i) s += smem[i];
    partials[blockIdx.x] = s;
  }
}

// Edge term: gather z[r], w[c], 16-dim distance, weighted sum.
__global__ __launch_bounds__(256) void edge_kernel(
    const float* __restrict__ g1, const float* __restrict__ g2,
    const float* __restrict__ z, const float* __restrict__ w,
    const float* __restrict__ gw, const int* __restrict__ ridx,
    const int* __restrict__ cidx, float* __restrict__ partials) {
  const int tid = blockIdx.x * 256 + threadIdx.x;
  const int stride = gridDim.x * 256;
  float acc = 0.f;
  for (int e = tid; e < Ev; e += stride) {
    const int r = ridx[e];
    const int c = cidx[e];
    const float4* zr = (const float4*)(z + (size_t)r * Dv);
    const float4* wr = (const float4*)(w + (size_t)c * Dv);
    float s = 0.f;
#pragma unroll
    for (int q = 0; q < 4; ++q) {
      float4 av = zr[q];
      float4 bv = wr[q];
      float dx = av.x - bv.x, dy = av.y - bv.y;
      float dz = av.z - bv.z, dw = av.w - bv.w;
      s += dx * dx + dy * dy + dz * dz + dw * dw;
    }
    acc += gw[e] * (g1[r] + g2[c] - DELTAv * __builtin_amdgcn_sqrtf(s));
  }
  const int lane = threadIdx.x & 31;
  const int wave = threadIdx.x >> 5;
#pragma unroll
  for (int off = 16; off > 0; off >>= 1) acc += __shfl_down(acc, off, 32);
  __shared__ float smem[8];
  if (lane == 0) smem[wave] = acc;
  __syncthreads();
  if (threadIdx.x == 0) {
    float s = 0.f;
#pragma unroll
    for (int i = 0; i < 8; ++i) s += smem[i];
    partials[blockIdx.x] = s;
  }
}

__global__ __launch_bounds__(256) void finalize_kernel(
    const float* __restrict__ p1, const float* __restrict__ p2,
    float* __restrict__ out) {
  __shared__ float smem[256];
  float s = 0.f;
  for (int i = threadIdx.x; i < 512; i += 256) s += p1[i] - p2[i];
  smem[threadIdx.x] = s;
  __syncthreads();
  for (int off = 128; off > 0; off >>= 1) {
    if (threadIdx.x < off) smem[threadIdx.x] += smem[threadIdx.x + off];
    __syncthreads();
  }
  if (threadIdx.x == 0) out[0] = smem[0];
}

extern "C" void kernel_launch(void* const* d_in, const int* in_sizes, int n_in,
                              void* d_out, int out_size, void* d_ws,
                              size_t ws_size, hipStream_t stream) {
  const float* g1 = (const float*)d_in[0];  // gamma_n1 [N1]
  const float* g2 = (const float*)d_in[1];  // gamma_n2 [N2]
  const float* z  = (const float*)d_in[2];  // latent_z [N1,16]
  const float* w  = (const float*)d_in[3];  // latent_w [N2,16]
  const float* gw = (const float*)d_in[4];  // graph_weights [E]
  const int* ridx = (const int*)d_in[5];    // rows_idx [E] (int32)
  const int* cidx = (const int*)d_in[6];    // col_idx  [E] (int32)
  float* out = (float*)d_out;
  float* ws = (float*)d_ws;

  (void)in_sizes; (void)n_in; (void)out_size; (void)ws_size;

  // 1) per-row norms + exp(gamma), column pairs interleaved
  precompute_kernel<<<(N1v + 255) / 256, 256, 0, stream>>>(g1, g2, z, w, ws);
  // 2) dense WMMA term -> 512 partials
  pdist1_kernel<<<N1v / 16, 256, 0, stream>>>(z, w, ws, ws + WS_P1);
  // 3) edge term -> 512 partials
  edge_kernel<<<512, 256, 0, stream>>>(g1, g2, z, w, gw, ridx, cidx, ws + WS_P2);
  // 4) deterministic final combine
  finalize_kernel<<<1, 256, 0, stream>>>(ws + WS_P1, ws + WS_P2, out);
}